// IsgnMeasNoteDecoder_25950192403223
// MI455X (gfx1250) — compile-verified
//
#include <hip/hip_runtime.h>
#include <math.h>

// ---------------- model dims ----------------
#define NNOTES   1024
#define NBEATS   128
#define NMEAS    32
#define ENC      128
#define ENCV     64
#define OUT_D    11
#define MARG     128
#define TREG     64
#define ETYPES   8
#define NHEAD    8
#define SEC      139              // OUT + MARGIN
#define NOTECAT  768
#define FINAL_IN 1165
#define BEAT_HID 898
#define RESDIM   8
#define MEASH    128

// ---------------- padded layout (all GEMM dims: M,N mult 64, K mult 32) ----------------
#define DPAD   1216               // FINAL_IN -> 19*64
#define LDX    (9*DPAD)           // owr block + 8 act blocks = 10944
#define SECP   144                // SEC padded (elementwise stride)
#define NGP    448                // fused z|r|h gate cols: z@0, r@144, h@288, pad@432 (7*64)
#define KINIT  800                // 770 -> mult 32
#define RXW    160                // r*x buffer width (K of uh GEMM), 5*32
#define HBW    192                // uh GEMM output width, 3*64

typedef __attribute__((ext_vector_type(2))) float v2f;
typedef __attribute__((ext_vector_type(4))) float v4f;
typedef __attribute__((ext_vector_type(8))) float v8f;

__device__ __forceinline__ float sigm(float x){ return 1.0f/(1.0f+expf(-x)); }

// =================== blocked fp32 WMMA GEMM ===================
// C[M x N] = A[M x K] @ B[K x N], row major. V_WMMA_F32_16X16X4_F32.
// Block = 4 waves = 64x64 C tile; wave w owns rows [16w,16w+16) x 64 cols
// (4 v8f accumulators). A fragment: one b64 global load reused by 4 WMMAs.
// B panel staged in LDS k-pair-interleaved: Bs2[p][c] = {B[2p][c], B[2p+1][c]},
// so one aligned ds_load_b64 yields a WMMA B operand with no register repacking.
// Pair-row stride 80 float2 => 32-bank rotation between adjacent pair rows =>
// lane halves hit disjoint bank sets; conflict-free.
// Requires M%64==0, N%64==0, K%32==0.
#define KC 32
__global__ void __launch_bounds__(128)
wmma_gemm_f32(const float* __restrict__ A, const float* __restrict__ B,
              float* __restrict__ C, int M, int N, int K,
              int lda, int ldb, int ldc)
{
    __shared__ v2f Bs2[KC / 2][80];
    int nb   = N >> 6;
    int bm   = blockIdx.x / nb;
    int bn   = blockIdx.x - bm * nb;
    int wave = threadIdx.x >> 5;
    int lane = threadIdx.x & 31;
    int half = lane >> 4;
    int l16  = lane & 15;
    int row0 = (bm << 6) + (wave << 4);
    int col0 = bn << 6;

    v8f acc[4];
#pragma unroll
    for (int j = 0; j < 4; j++)
#pragma unroll
        for (int v = 0; v < 8; v++) acc[j][v] = 0.0f;

    const float* Ap = A + (size_t)(row0 + l16) * lda + 2 * half;

    for (int k0 = 0; k0 < K; k0 += KC) {
        __syncthreads();
        // stage B[k0..k0+32) x [col0..col0+64) -> Bs2 (pair-interleaved)
#pragma unroll
        for (int it = 0; it < 2; it++) {
            int tIdx = threadIdx.x + it * 128;   // 0..255 (pair-row, col-group) tiles
            int q    = tIdx >> 4;                // pair row 0..15
            int cg   = (tIdx & 15) << 2;         // col group 0..60
            const float* s0 = B + (size_t)(k0 + 2 * q) * ldb + col0 + cg;
            v4f lo = *(const v4f*)s0;
            v4f hi = *(const v4f*)(s0 + ldb);
            if (k0 + KC < K) {
                __builtin_prefetch(s0 + (size_t)KC * ldb, 0, 1);
                __builtin_prefetch(s0 + (size_t)(KC + 1) * ldb, 0, 1);
            }
#pragma unroll
            for (int i = 0; i < 4; i++) {
                v2f pr; pr.x = lo[i]; pr.y = hi[i];
                Bs2[q][cg + i] = pr;
            }
        }
        __syncthreads();
        if (k0 + KC < K) __builtin_prefetch(Ap + k0 + KC, 0, 1);
#pragma unroll
        for (int kk = 0; kk < KC; kk += 4) {
            v2f a = *(const v2f*)(Ap + k0 + kk);
            int p = (kk >> 1) + half;            // this lane-half's k pair
#pragma unroll
            for (int j = 0; j < 4; j++) {
                v2f b = Bs2[p][(j << 4) + l16];
                acc[j] = __builtin_amdgcn_wmma_f32_16x16x4_f32(false, a, false, b,
                                                               (short)0, acc[j], false, false);
            }
        }
    }
#pragma unroll
    for (int j = 0; j < 4; j++)
#pragma unroll
        for (int v = 0; v < 8; v++)
            C[(size_t)(row0 + v + (half << 3)) * ldc + col0 + (j << 4) + l16] = acc[j][v];
}

// =================== weight prep ===================
__global__ void prep_wcat(const float* __restrict__ uz, const float* __restrict__ ur,
                          const float* __restrict__ wz, const float* __restrict__ wr,
                          const float* __restrict__ wh, float* __restrict__ Wc)
{
    size_t idx = (size_t)blockIdx.x * blockDim.x + threadIdx.x;
    if (idx >= (size_t)LDX * NGP) return;
    int r = (int)(idx / NGP), cc = (int)(idx % NGP);
    int grp = cc / SECP, d = cc - grp * SECP;
    float v = 0.0f;
    if (grp < 3 && d < SEC) {
        if (r < DPAD) {                      // u-part (applies to owr columns)
            int s = r;
            if (s < FINAL_IN) {
                if (grp == 0)      v = uz[s * SEC + d];
                else if (grp == 1) v = ur[s * SEC + d];
                // grp==2 (h): no dense u-term, uh is r-gated (separate GEMM)
            }
        } else {                             // w-part for act[e]
            int e = r / DPAD - 1;
            int s = r - (e + 1) * DPAD;
            if (s < FINAL_IN) {
                const float* w = (grp == 0) ? wz : (grp == 1) ? wr : wh;
                v = w[((size_t)e * FINAL_IN + s) * SEC + d];
            }
        }
    }
    Wc[idx] = v;
}

__global__ void prep_pad(const float* __restrict__ src, float* __restrict__ dst,
                         int sr, int sc, int dr, int dc)
{
    size_t idx = (size_t)blockIdx.x * blockDim.x + threadIdx.x;
    if (idx >= (size_t)dr * dc) return;
    int r = (int)(idx / dc), c = (int)(idx % dc);
    dst[idx] = (r < sr && c < sc) ? src[(size_t)r * sc + c] : 0.0f;
}

__global__ void transpose_edges(const float* __restrict__ E, float* __restrict__ ET)
{
    size_t idx = (size_t)blockIdx.x * blockDim.x + threadIdx.x;
    if (idx >= (size_t)ETYPES * NNOTES * NNOTES) return;
    size_t e = idx >> 20;
    int r = (int)((idx >> 10) & 1023);
    int c = (int)(idx & 1023);
    ET[idx] = E[(e << 20) | ((size_t)c << 10) | (size_t)r];
}

// =================== small dense kernels ===================
__global__ void style_k(const float* __restrict__ pe, const float* __restrict__ w,
                        const float* __restrict__ b, float* __restrict__ pz)
{
    int d = threadIdx.x;
    float a = b[d];
    for (int j = 0; j < ENCV; j++) a += pe[j] * w[j * ENC + d];
    pz[d] = fmaxf(a, 0.0f);
}

// 32-step measure decoder: two LSTM cells advanced in the same forward scan.
__global__ void __launch_bounds__(512)
measure_decode(const float* __restrict__ memb, const float* __restrict__ pz,
               const float* wi_f, const float* wh_f, const float* bi_f, const float* bh_f,
               const float* wi_b, const float* wh_b, const float* bi_b, const float* bh_b,
               const float* fcw, const float* fcb,
               float* __restrict__ mtv, float* __restrict__ dout_tail)
{
    int t = threadIdx.x;
    __shared__ float inp[386], hf[MEASH], cf[MEASH], hb[MEASH], cb[MEASH];
    __shared__ float gf[512], gb[512], prev[2];
    if (t < MEASH) { hf[t] = 0.f; cf[t] = 0.f; hb[t] = 0.f; cb[t] = 0.f; }
    if (t < 2) prev[t] = 0.f;
    __syncthreads();
    for (int m = 0; m < NMEAS; m++) {
        if (t < ENC)            inp[t] = pz[t];
        else if (t < ENC + 256) inp[t] = memb[m * 256 + (t - ENC)];
        if (t < 2)              inp[384 + t] = prev[t];
        __syncthreads();
        float af = bi_f[t] + bh_f[t];
        float ab = bi_b[t] + bh_b[t];
        for (int j = 0; j < 386; j++) {
            float x = inp[j];
            af += x * wi_f[j * 512 + t];
            ab += x * wi_b[j * 512 + t];
        }
        for (int j = 0; j < MEASH; j++) {
            af += hf[j] * wh_f[j * 512 + t];
            ab += hb[j] * wh_b[j * 512 + t];
        }
        gf[t] = af; gb[t] = ab;
        __syncthreads();
        if (t < MEASH) {
            float ig = sigm(gf[t]), fg = sigm(gf[MEASH + t]);
            float gg = tanhf(gf[2 * MEASH + t]), og = sigm(gf[3 * MEASH + t]);
            float cn = fg * cf[t] + ig * gg;
            cf[t] = cn; hf[t] = og * tanhf(cn);
        } else if (t < 2 * MEASH) {
            int u = t - MEASH;
            float ig = sigm(gb[u]), fg = sigm(gb[MEASH + u]);
            float gg = tanhf(gb[2 * MEASH + u]), og = sigm(gb[3 * MEASH + u]);
            float cn = fg * cb[u] + ig * gg;
            cb[u] = cn; hb[u] = og * tanhf(cn);
        }
        __syncthreads();
        if (t < 2) {
            float o = fcb[t];
            for (int j = 0; j < MEASH; j++) o += hf[j] * fcw[j * 2 + t];
            for (int j = 0; j < MEASH; j++) o += hb[j] * fcw[(MEASH + j) * 2 + t];
            prev[t] = o; mtv[m * 2 + t] = o; dout_tail[m * 2 + t] = o;
        }
        __syncthreads();
    }
}

__global__ void build_note(const float* __restrict__ tnc, const float* __restrict__ mtv,
                           const float* __restrict__ pz, const int* __restrict__ measn,
                           float* __restrict__ X)
{
    int n = blockIdx.x;
    int ms = measn[n] - measn[0];
    float* row = X + (size_t)n * LDX;
    for (int c = threadIdx.x; c < NOTECAT; c += blockDim.x) row[c] = tnc[(size_t)n * NOTECAT + c];
    if (threadIdx.x < 2) row[NOTECAT + threadIdx.x] = mtv[ms * 2 + threadIdx.x];
    for (int c = threadIdx.x; c < ENC; c += blockDim.x) row[770 + c] = pz[c];
}

__global__ void init_fc2_k(const float* __restrict__ hid, const float* __restrict__ b1,
                           const float* __restrict__ w2, const float* __restrict__ b2,
                           float* __restrict__ X)
{
    int n = blockIdx.x, d = threadIdx.x;
    __shared__ float sh[128];
    for (int j = d; j < 128; j += 32) sh[j] = fmaxf(hid[(size_t)n * 128 + j] + b1[j], 0.0f);
    __syncthreads();
    if (d < OUT_D) {
        float a = b2[d];
        for (int j = 0; j < 128; j++) a += sh[j] * w2[j * OUT_D + d];
        X[(size_t)n * LDX + 1026 + d] = a;
    }
}

// =================== gated graph elementwise ===================
__global__ void graph_elem_a(const float* __restrict__ pre, const float* __restrict__ bz,
                             const float* __restrict__ br, const float* __restrict__ X,
                             float* __restrict__ zb, float* __restrict__ rx)
{
    int n = blockIdx.x, d = threadIdx.x;   // 160 threads
    float zv = 0.f, rv = 0.f;
    if (d < SEC) {
        zv = sigm(pre[(size_t)n * NGP + d] + bz[d]);
        float r = sigm(pre[(size_t)n * NGP + SECP + d] + br[d]);
        rv = r * X[(size_t)n * LDX + 1026 + d];
    }
    if (d < SECP) zb[(size_t)n * SECP + d] = zv;
    rx[(size_t)n * RXW + d] = rv;
}

__global__ void graph_elem_b(const float* __restrict__ pre, const float* __restrict__ hbuf,
                             const float* __restrict__ bh, const float* __restrict__ zb,
                             float* __restrict__ X)
{
    int n = blockIdx.x, d = threadIdx.x;   // 144 threads
    if (d < SEC) {
        float h = tanhf(pre[(size_t)n * NGP + 2 * SECP + d] + hbuf[(size_t)n * HBW + d] + bh[d]);
        float z = zb[(size_t)n * SECP + d];
        size_t ix = (size_t)n * LDX + 1026 + d;
        float xs = X[ix];
        X[ix] = (1.0f - z) * xs + z * h;
    }
}

// =================== segment attention ===================
__global__ void margin_att_note(const float* __restrict__ X, const float* __restrict__ W,
                                const float* __restrict__ b, const float* __restrict__ ctx,
                                float* __restrict__ sim)
{
    int n = blockIdx.x, d = threadIdx.x;   // 128
    __shared__ float xr[MARG], at[MARG];
    xr[d] = X[(size_t)n * LDX + (FINAL_IN - MARG) + d];
    __syncthreads();
    float a = b[d];
    for (int j = 0; j < MARG; j++) a += xr[j] * W[j * MARG + d];
    at[d] = tanhf(a);
    __syncthreads();
    if (d < NHEAD) {
        float s = 0.f;
        for (int k = 0; k < 16; k++) s += at[d * 16 + k] * ctx[d * 16 + k];
        sim[n * NHEAD + d] = s;
    }
}

__global__ void margin_att_beat(const float* __restrict__ X, const float* __restrict__ sim,
                                float* __restrict__ xbeat)
{
    int bt = blockIdx.x, d = threadIdx.x;  // 128
    __shared__ float w[NHEAD * 8];
    if (d < NHEAD) {
        float m = -1e30f;
        for (int i = 0; i < 8; i++) m = fmaxf(m, sim[(bt * 8 + i) * NHEAD + d]);
        float e[8], s = 0.f;
        for (int i = 0; i < 8; i++) { e[i] = expf(sim[(bt * 8 + i) * NHEAD + d] - m); s += e[i]; }
        for (int i = 0; i < 8; i++) w[d * 8 + i] = e[i] / s;
    }
    __syncthreads();
    int h = d >> 4;
    float acc = 0.f;
    for (int i = 0; i < 8; i++)
        acc += X[(size_t)(bt * 8 + i) * LDX + (FINAL_IN - MARG) + d] * w[h * 8 + i];
    xbeat[bt * 147 + OUT_D + d] = acc;
}

__global__ void beat_att_note(const float* __restrict__ X, const float* __restrict__ W,
                              const float* __restrict__ b, const float* __restrict__ ctx,
                              float* __restrict__ sim)
{
    int n = blockIdx.x, d = threadIdx.x;   // 32
    __shared__ float xr[OUT_D], at[OUT_D];
    if (d < OUT_D) xr[d] = X[(size_t)n * LDX + 1026 + d];
    __syncthreads();
    if (d < OUT_D) {
        float a = b[d];
        for (int j = 0; j < OUT_D; j++) a += xr[j] * W[j * OUT_D + d];
        at[d] = tanhf(a);
    }
    __syncthreads();
    if (d == 0) {
        float s = 0.f;
        for (int k = 0; k < OUT_D; k++) s += at[k] * ctx[k];
        sim[n] = s;
    }
}

__global__ void beat_att_beat(const float* __restrict__ X, const float* __restrict__ sim,
                              const float* __restrict__ res, float* __restrict__ xbeat)
{
    int bt = blockIdx.x, d = threadIdx.x;  // 16
    __shared__ float w[8];
    if (d == 0) {
        float m = -1e30f;
        for (int i = 0; i < 8; i++) m = fmaxf(m, sim[bt * 8 + i]);
        float e[8], s = 0.f;
        for (int i = 0; i < 8; i++) { e[i] = expf(sim[bt * 8 + i] - m); s += e[i]; }
        for (int i = 0; i < 8; i++) w[i] = e[i] / s;
    }
    __syncthreads();
    if (d < OUT_D) {
        float acc = 0.f;
        for (int i = 0; i < 8; i++) acc += X[(size_t)(bt * 8 + i) * LDX + 1026 + d] * w[i];
        xbeat[bt * 147 + d] = acc;
    }
    if (d < RESDIM) xbeat[bt * 147 + OUT_D + MARG + d] = res[bt * RESDIM + d];
}

// =================== beat bi-LSTM layer (block 0 = fwd, 1 = bwd) ===================
__global__ void __launch_bounds__(256)
bilstm_layer(const float* __restrict__ xs, int din,
             const float* wi_f, const float* wh_f, const float* bi_f, const float* bh_f,
             const float* wi_b, const float* wh_b, const float* bi_b, const float* bh_b,
             float* __restrict__ out)
{
    int dir = blockIdx.x;
    const float* wi = dir ? wi_b : wi_f;
    const float* wh = dir ? wh_b : wh_f;
    const float* bi = dir ? bi_b : bi_f;
    const float* bh = dir ? bh_b : bh_f;
    int t = threadIdx.x;
    __shared__ float h[TREG], c[TREG], g[4 * TREG], xr[160];
    if (t < TREG) { h[t] = 0.f; c[t] = 0.f; }
    __syncthreads();
    for (int step = 0; step < NBEATS; step++) {
        int s = dir ? (NBEATS - 1 - step) : step;
        if (t < din) xr[t] = xs[s * din + t];
        __syncthreads();
        float a = bi[t] + bh[t];
        for (int j = 0; j < din; j++) a += xr[j] * wi[j * 256 + t];
        for (int j = 0; j < TREG; j++) a += h[j] * wh[j * 256 + t];
        g[t] = a;
        __syncthreads();
        if (t < TREG) {
            float ig = sigm(g[t]), fg = sigm(g[TREG + t]);
            float gg = tanhf(g[2 * TREG + t]), og = sigm(g[3 * TREG + t]);
            float cn = fg * c[t] + ig * gg;
            float hn = og * tanhf(cn);
            c[t] = cn; h[t] = hn;
            out[s * (2 * TREG) + dir * TREG + t] = hn;
        }
        __syncthreads();
    }
}

__global__ void tempo_k(const float* __restrict__ brnn, const float* __restrict__ w,
                        const float* __restrict__ b, float* __restrict__ tpo)
{
    int bt = threadIdx.x;  // 128
    float a = b[0];
    for (int j = 0; j < 128; j++) a += brnn[bt * 128 + j] * w[j];
    tpo[bt] = a;
}

__global__ void spread_beat(const float* __restrict__ brnn, const int* __restrict__ beatn,
                            float* __restrict__ X)
{
    int n = blockIdx.x, d = threadIdx.x;   // 128
    int sg = beatn[n] - beatn[0];
    X[(size_t)n * LDX + BEAT_HID + d] = brnn[sg * 128 + d];
}

__global__ void fc2_final_k(const float* __restrict__ hid, const float* __restrict__ b1,
                            const float* __restrict__ w2, const float* __restrict__ b2,
                            const float* __restrict__ tpo, const int* __restrict__ beatn,
                            float* __restrict__ X, float* __restrict__ dout)
{
    int n = blockIdx.x, d = threadIdx.x;   // 32
    __shared__ float sh[128];
    for (int j = d; j < 128; j += 32) sh[j] = fmaxf(hid[(size_t)n * 128 + j] + b1[j], 0.0f);
    __syncthreads();
    int sg = beatn[n] - beatn[0];
    if (d == 0) {
        float tv = tpo[sg];
        X[(size_t)n * LDX + 1026] = tv;
        dout[n * OUT_D] = tv;
    }
    if (d < 10) {
        float a = b2[d];
        for (int j = 0; j < 128; j++) a += sh[j] * w2[j * 10 + d];
        X[(size_t)n * LDX + 1027 + d] = a;
        dout[n * OUT_D + 1 + d] = a;
    }
}

// =================== host launcher ===================
extern "C" void kernel_launch(void* const* d_in, const int* in_sizes, int n_in,
                              void* d_out, int out_size, void* d_ws, size_t ws_size,
                              hipStream_t stream)
{
    (void)in_sizes; (void)n_in; (void)out_size; (void)ws_size;
    // inputs (setup_inputs order), then params pytree leaves (dicts sorted by key)
    const float* tnc   = (const float*)d_in[0];
    const float* memb  = (const float*)d_in[1];
    const float* perf  = (const float*)d_in[2];
    const float* res   = (const float*)d_in[3];
    const float* edges = (const float*)d_in[4];
    const int*   beatn = (const int*)d_in[5];
    const int*   measn = (const int*)d_in[6];
    const float* ba_b   = (const float*)d_in[7];
    const float* ba_ctx = (const float*)d_in[8];
    const float* ba_w   = (const float*)d_in[9];
    const float* fc1_b  = (const float*)d_in[10];
    const float* fc1_w  = (const float*)d_in[11];
    const float* fc2_b  = (const float*)d_in[12];
    const float* fc2_w  = (const float*)d_in[13];
    const float* gg_bh  = (const float*)d_in[14];
    const float* gg_br  = (const float*)d_in[15];
    const float* gg_bz  = (const float*)d_in[16];
    const float* gg_uh  = (const float*)d_in[17];
    const float* gg_ur  = (const float*)d_in[18];
    const float* gg_uz  = (const float*)d_in[19];
    const float* gg_wh  = (const float*)d_in[20];
    const float* gg_wr  = (const float*)d_in[21];
    const float* gg_wz  = (const float*)d_in[22];
    const float* if1_b  = (const float*)d_in[23];
    const float* if1_w  = (const float*)d_in[24];
    const float* if2_b  = (const float*)d_in[25];
    const float* if2_w  = (const float*)d_in[26];
    const float* ma_b   = (const float*)d_in[27];
    const float* ma_ctx = (const float*)d_in[28];
    const float* ma_w   = (const float*)d_in[29];
    const float* me_bwd_bh = (const float*)d_in[30];
    const float* me_bwd_bi = (const float*)d_in[31];
    const float* me_bwd_wh = (const float*)d_in[32];
    const float* me_bwd_wi = (const float*)d_in[33];
    const float* me_fc_b   = (const float*)d_in[34];
    const float* me_fc_w   = (const float*)d_in[35];
    const float* me_fwd_bh = (const float*)d_in[36];
    const float* me_fwd_bi = (const float*)d_in[37];
    const float* me_fwd_wh = (const float*)d_in[38];
    const float* me_fwd_wi = (const float*)d_in[39];
    const float* st_b = (const float*)d_in[40];
    const float* st_w = (const float*)d_in[41];
    const float* tf_b = (const float*)d_in[42];
    const float* tf_w = (const float*)d_in[43];
    const float* t0b_bh = (const float*)d_in[44];
    const float* t0b_bi = (const float*)d_in[45];
    const float* t0b_wh = (const float*)d_in[46];
    const float* t0b_wi = (const float*)d_in[47];
    const float* t0f_bh = (const float*)d_in[48];
    const float* t0f_bi = (const float*)d_in[49];
    const float* t0f_wh = (const float*)d_in[50];
    const float* t0f_wi = (const float*)d_in[51];
    const float* t1b_bh = (const float*)d_in[52];
    const float* t1b_bi = (const float*)d_in[53];
    const float* t1b_wh = (const float*)d_in[54];
    const float* t1b_wi = (const float*)d_in[55];
    const float* t1f_bh = (const float*)d_in[56];
    const float* t1f_bi = (const float*)d_in[57];
    const float* t1f_wh = (const float*)d_in[58];
    const float* t1f_wi = (const float*)d_in[59];

    // workspace layout (floats)
    float* W = (float*)d_ws;
    size_t o = 0;
    auto alloc = [&](size_t n) { float* p = W + o; o += n; return p; };
    float* X    = alloc((size_t)NNOTES * LDX);          // owr + 8 act blocks (zero padded)
    float* Wcat = alloc((size_t)LDX * NGP);             // [uz|ur|0|0 ; wz|wr|wh|0] fused
    float* ET   = alloc((size_t)ETYPES * NNOTES * NNOTES);
    float* pre  = alloc((size_t)NNOTES * NGP);
    float* zb   = alloc((size_t)NNOTES * SECP);
    float* rx   = alloc((size_t)NNOTES * RXW);
    float* hbuf = alloc((size_t)NNOTES * HBW);
    float* hid  = alloc((size_t)NNOTES * 128);
    float* uhp  = alloc((size_t)RXW * HBW);
    float* w1ip = alloc((size_t)KINIT * 128);
    float* wf1p = alloc((size_t)DPAD * 128);
    float* pz   = alloc(ENC);
    float* mtv  = alloc(NMEAS * 2);
    float* simM = alloc((size_t)NNOTES * NHEAD);
    float* simB = alloc(NNOTES);
    float* xbeat= alloc((size_t)NBEATS * 147);
    float* l1o  = alloc((size_t)NBEATS * 128);
    float* brnn = alloc((size_t)NBEATS * 128);
    float* tpo  = alloc(NBEATS);
    float* fout = (float*)d_out;

    // X must be clean every call (zero pads, zero beat-hidden, zero margin)
    hipMemsetAsync(X, 0, (size_t)NNOTES * LDX * sizeof(float), stream);

    { size_t tot = (size_t)LDX * NGP;
      prep_wcat<<<(unsigned)((tot + 255) / 256), 256, 0, stream>>>(gg_uz, gg_ur, gg_wz, gg_wr, gg_wh, Wcat); }
    prep_pad<<<(RXW * HBW + 255) / 256, 256, 0, stream>>>(gg_uh, uhp, SEC, SEC, RXW, HBW);
    prep_pad<<<(KINIT * 128 + 255) / 256, 256, 0, stream>>>(if1_w, w1ip, 770, 128, KINIT, 128);
    prep_pad<<<(DPAD * 128 + 255) / 256, 256, 0, stream>>>(fc1_w, wf1p, FINAL_IN, 128, DPAD, 128);
    { size_t tot = (size_t)ETYPES * NNOTES * NNOTES;
      transpose_edges<<<(unsigned)((tot + 255) / 256), 256, 0, stream>>>(edges, ET); }

    style_k<<<1, ENC, 0, stream>>>(perf, st_w, st_b, pz);
    measure_decode<<<1, 512, 0, stream>>>(memb, pz,
        me_fwd_wi, me_fwd_wh, me_fwd_bi, me_fwd_bh,
        me_bwd_wi, me_bwd_wh, me_bwd_bi, me_bwd_bh,
        me_fc_w, me_fc_b, mtv, fout + NNOTES * OUT_D);
    build_note<<<NNOTES, 256, 0, stream>>>(tnc, mtv, pz, measn, X);

    auto gemm = [&](const float* A, const float* B, float* C,
                    int M, int N, int K, int lda, int ldb, int ldc) {
        int blocks = (M >> 6) * (N >> 6);
        wmma_gemm_f32<<<blocks, 128, 0, stream>>>(A, B, C, M, N, K, lda, ldb, ldc);
    };

    // init_out = fc2(relu(fc1(note_out)))
    gemm(X, w1ip, hid, NNOTES, 128, KINIT, LDX, 128, 128);
    init_fc2_k<<<NNOTES, 32, 0, stream>>>(hid, if1_b, if2_w, if2_b, X);

    for (int si = 0; si < 2; si++) {
        for (int gi = 0; gi < 2; gi++) {
            // act[e] = edges[e]^T @ owr  (edges pre-transposed)
            for (int e = 0; e < ETYPES; e++)
                gemm(ET + (size_t)e * NNOTES * NNOTES, X, X + (size_t)(e + 1) * DPAD,
                     NNOTES, DPAD, NNOTES, NNOTES, LDX, LDX);
            // fused [z|r|h] pre-activations: (owr ++ act[0..7]) @ Wcat
            gemm(X, Wcat, pre, NNOTES, NGP, LDX, LDX, NGP, NGP);
            graph_elem_a<<<NNOTES, RXW, 0, stream>>>(pre, gg_bz, gg_br, X, zb, rx);
            // h_gate += (r * x_sec) @ uh  (separate buffer, zero-padded weights)
            gemm(rx, uhp, hbuf, NNOTES, HBW, RXW, RXW, HBW, HBW);
            graph_elem_b<<<NNOTES, SECP, 0, stream>>>(pre, hbuf, gg_bh, zb, X);
        }
        margin_att_note<<<NNOTES, MARG, 0, stream>>>(X, ma_w, ma_b, ma_ctx, simM);
        margin_att_beat<<<NBEATS, MARG, 0, stream>>>(X, simM, xbeat);
        beat_att_note<<<NNOTES, 32, 0, stream>>>(X, ba_w, ba_b, ba_ctx, simB);
        beat_att_beat<<<NBEATS, 16, 0, stream>>>(X, simB, res, xbeat);
        bilstm_layer<<<2, 256, 0, stream>>>(xbeat, 147,
            t0f_wi, t0f_wh, t0f_bi, t0f_bh, t0b_wi, t0b_wh, t0b_bi, t0b_bh, l1o);
        bilstm_layer<<<2, 256, 0, stream>>>(l1o, 128,
            t1f_wi, t1f_wh, t1f_bi, t1f_bh, t1b_wi, t1b_wh, t1b_bi, t1b_bh, brnn);
        tempo_k<<<1, NBEATS, 0, stream>>>(brnn, tf_w, tf_b, tpo);
        spread_beat<<<NNOTES, MARG, 0, stream>>>(brnn, beatn, X);
        gemm(X, wf1p, hid, NNOTES, 128, DPAD, LDX, 128, 128);
        fc2_final_k<<<NNOTES, 32, 0, stream>>>(hid, fc1_b, fc2_w, fc2_b, tpo, beatn, X, fout);
    }
}